// SquatGNN_45999099740722
// MI455X (gfx1250) — compile-verified
//
#include <hip/hip_runtime.h>
#include <hip/hip_bf16.h>

typedef __attribute__((ext_vector_type(2))) float v2f;
typedef __attribute__((ext_vector_type(8))) float v8f;

// ---------------------------------------------------------------------------
// Degree / normalization
// ---------------------------------------------------------------------------
__global__ void deg_init_kernel(float* __restrict__ deg, int n) {
    int i = blockIdx.x * blockDim.x + threadIdx.x;
    if (i < n) deg[i] = 1.0f;  // self loop contributes 1
}

__global__ void deg_scatter_kernel(const int* __restrict__ dst,
                                   float* __restrict__ deg, int E) {
    int e = blockIdx.x * blockDim.x + threadIdx.x;
    if (e < E) atomicAdd(&deg[dst[e]], 1.0f);
}

__global__ void deg_rsqrt_kernel(float* __restrict__ deg, int n) {
    int i = blockIdx.x * blockDim.x + threadIdx.x;
    if (i < n) {
        float d = deg[i];
        deg[i] = (d > 0.0f) ? rsqrtf(d) : 0.0f;  // becomes dinv in place
    }
}

// ---------------------------------------------------------------------------
// Dense GEMM H[M,N] = act(X[M,K]) @ W[K,N] via V_WMMA_F32_16X16X4_F32.
//
// Block <-> one 16-row M-tile; wave w <-> N-tile n0 = 16*w (blockDim = N/16
// waves). The block cooperatively stages the FULL W (K*N floats, <=64KB) and
// its 16 A-rows (16*K floats, <=8KB, activation fused) into dynamic LDS, then
// each wave runs the K-loop entirely out of LDS. gfx1250 WGPs own 320KB LDS,
// so the 72KB worst case still allows multiple blocks per WGP.
//
// Fragment layouts (CDNA5 ISA §7.12.2, 32-bit):
//   A 16x4 : lanes 0-15 rows, {K=k0,k0+1}; lanes 16-31 same rows {k0+2,k0+3}
//   B 4x16 : lanes 0-15 cols, rows {k0,k0+1}; lanes 16-31 rows {k0+2,k0+3}
//   C 16x16: VGPR r -> lanes 0-15 row m0+r, lanes 16-31 row m0+8+r
// ---------------------------------------------------------------------------
template <int RELU>
__global__ void gemm_wmma_f32_kernel(const float* __restrict__ X,
                                     const float* __restrict__ W,
                                     float* __restrict__ H,
                                     int M, int K, int N) {
    extern __shared__ float lds[];
    float* Wl = lds;                       // [K][N]
    float* Al = lds + (size_t)K * N;       // [16][K]

    const int t  = threadIdx.x;
    const int nt = blockDim.x;
    const int m0 = blockIdx.x << 4;

    // Stage W (float4 cooperative copy, fully coalesced).
    const int wn4 = (K * N) >> 2;
    for (int i = t; i < wn4; i += nt)
        ((float4*)Wl)[i] = ((const float4*)W)[i];

    // Stage 16 A rows, fusing the input activation; clamp row for partial tiles.
    const int k4  = K >> 2;
    const int an4 = 16 * k4;
    for (int i = t; i < an4; i += nt) {
        int r   = i / k4;
        int c4  = i % k4;
        int row = m0 + r;
        if (row > M - 1) row = M - 1;
        float4 v = ((const float4*)(X + (size_t)row * K))[c4];
        if (RELU) {
            v.x = fmaxf(v.x, 0.0f);
            v.y = fmaxf(v.y, 0.0f);
            v.z = fmaxf(v.z, 0.0f);
            v.w = fmaxf(v.w, 0.0f);
        }
        ((float4*)Al)[i] = v;
    }
    __syncthreads();

    const int lane = t & 31;
    const int wave = t >> 5;               // == N-tile index
    const int n0   = wave << 4;
    const int half = lane >> 4;            // 0: K pair {0,1}; 1: {2,3}
    const int l    = lane & 15;

    v8f acc = {};
    for (int k0 = 0; k0 < K; k0 += 4) {
        const int ka = k0 + (half << 1);
        v2f a, b;
        a.x = Al[l * K + ka];
        a.y = Al[l * K + ka + 1];
        b.x = Wl[ka * N + n0 + l];
        b.y = Wl[(ka + 1) * N + n0 + l];
        acc = __builtin_amdgcn_wmma_f32_16x16x4_f32(
            /*neg_a=*/false, a, /*neg_b=*/false, b,
            /*c_mod=*/(short)0, acc, /*reuse_a=*/false, /*reuse_b=*/false);
    }

    const int mrow = m0 + half * 8;
#pragma unroll
    for (int r = 0; r < 8; ++r) {
        int row = mrow + r;
        if (row < M) H[(size_t)row * N + n0 + l] = acc[r];
    }
}

// ---------------------------------------------------------------------------
// AGG[n][c] = bias[c] + H[n][c] * dinv[n]^2   (bias + self-loop term)
// ---------------------------------------------------------------------------
__global__ void agg_init_kernel(const float* __restrict__ H,
                                const float* __restrict__ dinv,
                                const float* __restrict__ bias,
                                float* __restrict__ AGG,
                                int Nn, int F4) {   // F4 = F/4
    long long i4 = (long long)blockIdx.x * blockDim.x + threadIdx.x;
    long long total = (long long)Nn * F4;
    if (i4 >= total) return;
    int n  = (int)(i4 / F4);
    int c4 = (int)(i4 % F4);
    float di = dinv[n];
    float w  = di * di;
    float4 h = ((const float4*)H)[i4];
    float4 b = ((const float4*)bias)[c4];
    float4 o;
    o.x = b.x + h.x * w;
    o.y = b.y + h.y * w;
    o.z = b.z + h.z * w;
    o.w = b.w + h.w * w;
    ((float4*)AGG)[i4] = o;
}

// ---------------------------------------------------------------------------
// Edge scatter: one wave32 per edge; 128 floats = 4 per lane (float4 gather,
// 4 f32 atomic adds -> 512B contiguous per wave). This is the memory-bound
// dominant phase (~6.5GB total at 23.3TB/s HBM).
// ---------------------------------------------------------------------------
__global__ void edge_scatter_kernel(const float* __restrict__ H,
                                    const float* __restrict__ dinv,
                                    const int* __restrict__ src,
                                    const int* __restrict__ dst,
                                    float* __restrict__ AGG,
                                    int E, int F) {
    const int lane = threadIdx.x & 31;
    long long e = (long long)blockIdx.x * (blockDim.x >> 5) + (threadIdx.x >> 5);
    if (e >= E) return;
    const int s = src[e];
    const int d = dst[e];
    const float nrm = dinv[s] * dinv[d];

    const float4 v = ((const float4*)(H + (size_t)s * F))[lane];
    float* ap = AGG + (size_t)d * F + lane * 4;
    atomicAdd(ap + 0, v.x * nrm);
    atomicAdd(ap + 1, v.y * nrm);
    atomicAdd(ap + 2, v.z * nrm);
    atomicAdd(ap + 3, v.w * nrm);
}

// ---------------------------------------------------------------------------
// Pooling: zero sums/cnts, then scatter relu(AGG2) per node into graph bins.
// ---------------------------------------------------------------------------
__global__ void zero_kernel(float* __restrict__ p, long long n) {
    long long i = (long long)blockIdx.x * blockDim.x + threadIdx.x;
    if (i < n) p[i] = 0.0f;
}

__global__ void pool_scatter_kernel(const float* __restrict__ AGG,
                                    const int* __restrict__ batch,
                                    float* __restrict__ sums,
                                    float* __restrict__ cnts,
                                    int Nn, int F) {
    const int lane = threadIdx.x & 31;
    long long n = (long long)blockIdx.x * (blockDim.x >> 5) + (threadIdx.x >> 5);
    if (n >= Nn) return;
    const int g = batch[n];
    float4 v = ((const float4*)(AGG + (size_t)n * F))[lane];
    v.x = fmaxf(v.x, 0.0f);
    v.y = fmaxf(v.y, 0.0f);
    v.z = fmaxf(v.z, 0.0f);
    v.w = fmaxf(v.w, 0.0f);
    float* sp = sums + (size_t)g * F + lane * 4;
    atomicAdd(sp + 0, v.x);
    atomicAdd(sp + 1, v.y);
    atomicAdd(sp + 2, v.z);
    atomicAdd(sp + 3, v.w);
    if (lane == 0) atomicAdd(&cnts[g], 1.0f);
}

// ---------------------------------------------------------------------------
// Head: pooled = sums / max(cnt,1); logits = pooled @ Wfc + bfc; log_softmax.
// One thread per graph (tiny: 64 x 128 x 2).
// ---------------------------------------------------------------------------
__global__ void head_kernel(const float* __restrict__ sums,
                            const float* __restrict__ cnts,
                            const float* __restrict__ Wfc,
                            const float* __restrict__ bfc,
                            float* __restrict__ out,
                            int G, int F, int O) {
    int g = blockIdx.x * blockDim.x + threadIdx.x;
    if (g >= G) return;
    float inv = 1.0f / fmaxf(cnts[g], 1.0f);
    float logits[8];
    for (int o = 0; o < O; ++o) {
        float acc = bfc[o];
        for (int k = 0; k < F; ++k)
            acc += (sums[(size_t)g * F + k] * inv) * Wfc[(size_t)k * O + o];
        logits[o] = acc;
    }
    float mx = logits[0];
    for (int o = 1; o < O; ++o) mx = fmaxf(mx, logits[o]);
    float se = 0.0f;
    for (int o = 0; o < O; ++o) se += expf(logits[o] - mx);
    float lse = mx + logf(se);
    for (int o = 0; o < O; ++o) out[(size_t)g * O + o] = logits[o] - lse;
}

// ---------------------------------------------------------------------------
// Launch
// ---------------------------------------------------------------------------
extern "C" void kernel_launch(void* const* d_in, const int* in_sizes, int n_in,
                              void* d_out, int out_size, void* d_ws, size_t ws_size,
                              hipStream_t stream) {
    const float* x    = (const float*)d_in[0];
    const float* W1   = (const float*)d_in[1];
    const float* b1   = (const float*)d_in[2];
    const float* W2   = (const float*)d_in[3];
    const float* b2   = (const float*)d_in[4];
    const float* Wfc  = (const float*)d_in[5];
    const float* bfc  = (const float*)d_in[6];
    const int*   ei   = (const int*)d_in[7];   // [2, E] flattened
    const int*   batch= (const int*)d_in[8];

    const int Nn  = in_sizes[8];           // 100000 nodes
    const int IND = in_sizes[0] / Nn;      // 64
    const int HID = in_sizes[2];           // 128
    const int E   = in_sizes[7] / 2;       // 1.6M
    const int O   = in_sizes[6];           // 2
    const int G   = out_size / O;          // 64

    const int* src = ei;
    const int* dst = ei + E;

    // Workspace layout (floats)
    size_t NH = (size_t)Nn * HID;
    float* h    = (float*)d_ws;            // [Nn, HID]
    float* agg  = h + NH;                  // [Nn, HID]
    float* dinv = agg + NH;                // [Nn]   (deg -> rsqrt in place)
    float* sums = dinv + Nn;               // [G, HID]
    float* cnts = sums + (size_t)G * HID;  // [G]
    float* outf = (float*)d_out;

    const int BT = 256;
    auto cdiv = [](long long a, long long b) { return (int)((a + b - 1) / b); };

    // --- symmetric degree normalization ---
    deg_init_kernel   <<<cdiv(Nn, BT), BT, 0, stream>>>(dinv, Nn);
    deg_scatter_kernel<<<cdiv(E,  BT), BT, 0, stream>>>(dst, dinv, E);
    deg_rsqrt_kernel  <<<cdiv(Nn, BT), BT, 0, stream>>>(dinv, Nn);

    // GEMM launcher: 1 block per 16-row M-tile; N/16 waves per block;
    // dynamic LDS = full W + 16 A rows.
    auto launch_gemm = [&](const float* Xp, const float* Wm, float* Hh,
                           int M, int K, int Nc, int relu_in) {
        int blocks  = (M + 15) / 16;
        int threads = (Nc / 16) * 32;
        size_t shmem = ((size_t)K * Nc + 16 * (size_t)K) * sizeof(float);
        if (relu_in)
            gemm_wmma_f32_kernel<1><<<blocks, threads, shmem, stream>>>(
                Xp, Wm, Hh, M, K, Nc);
        else
            gemm_wmma_f32_kernel<0><<<blocks, threads, shmem, stream>>>(
                Xp, Wm, Hh, M, K, Nc);
    };

    const int F4 = HID / 4;
    const int waves_per_block = BT / 32;

    // --- layer 1: h = x @ W1 ; agg = b1 + self + scatter ---
    launch_gemm(x, W1, h, Nn, IND, HID, /*relu_in=*/0);
    agg_init_kernel<<<cdiv((long long)Nn * F4, BT), BT, 0, stream>>>(
        h, dinv, b1, agg, Nn, F4);
    edge_scatter_kernel<<<cdiv(E, waves_per_block), BT, 0, stream>>>(
        h, dinv, src, dst, agg, E, HID);

    // --- layer 2: h = relu(agg) @ W2 ; agg = b2 + self + scatter ---
    launch_gemm(agg, W2, h, Nn, HID, HID, /*relu_in=*/1);
    agg_init_kernel<<<cdiv((long long)Nn * F4, BT), BT, 0, stream>>>(
        h, dinv, b2, agg, Nn, F4);
    edge_scatter_kernel<<<cdiv(E, waves_per_block), BT, 0, stream>>>(
        h, dinv, src, dst, agg, E, HID);

    // --- global mean pool (relu fused into load) + FC head + log_softmax ---
    long long pool_elems = (long long)G * HID + G;
    zero_kernel<<<cdiv(pool_elems, BT), BT, 0, stream>>>(sums, pool_elems);
    pool_scatter_kernel<<<cdiv(Nn, waves_per_block), BT, 0, stream>>>(
        agg, batch, sums, cnts, Nn, HID);
    head_kernel<<<cdiv(G, 64), 64, 0, stream>>>(sums, cnts, Wfc, bfc, outf,
                                                G, HID, O);
}